// PatchletsExtractor_12781822673295
// MI455X (gfx1250) — compile-verified
//
#include <hip/hip_runtime.h>

typedef __attribute__((ext_vector_type(2))) float v2f;
typedef __attribute__((ext_vector_type(4))) float v4f;
typedef __attribute__((ext_vector_type(8))) float v8f;

#define BB  4
#define TT  32
#define NN  2048
#define KNN 16

// d_out float offsets (tuple outputs concatenated flat, reference return order)
#define O_IDX  0L
#define O_DST  (O_IDX + (long)BB*TT*NN*KNN)        // distances
#define O_PLT  (O_DST + (long)BB*TT*NN*KNN)        // patchlets (== idxs)
#define O_PTS  (O_PLT + (long)BB*TT*NN*KNN)        // patchlet_points  (..,K,3)
#define O_FTS  (O_PTS + (long)BB*TT*NN*KNN*3)      // patchlet_feats   (..,K,6)
#define O_NRM  (O_FTS + (long)BB*TT*NN*KNN*6)      // normalized pts   (..,K,3)
#define O_OUTX (O_NRM + (long)BB*TT*NN*KNN*3)      // out_x            (..,N,3)

// Copy point_seq[:,0,:,:] into the x_cur workspace (re-done every call: deterministic).
__global__ void patchlet_init(const float* __restrict__ pts, float* __restrict__ xcur) {
  int i = blockIdx.x * blockDim.x + threadIdx.x;
  if (i < BB * NN * 3) {
    int b = i / (NN * 3);
    int r = i - b * (NN * 3);
    xcur[i] = pts[(long)b * TT * NN * 3 + r];
  }
}

// One scan step: fused distance (WMMA f32 16x16x4) + top-16 + all gathers/outputs.
// Grid: BB * (NN/256) blocks of 256 threads (8 wave32; each wave owns 32 queries).
__global__ __launch_bounds__(256) void patchlet_step(const float* __restrict__ pts,
                                                     float* __restrict__ xcur,
                                                     float* __restrict__ out,
                                                     int t) {
  __shared__ float sref[NN * 3];       // all refs for (b, t-1): 24 KB
  __shared__ float sqc[256 * 3];       // this block's query coords
  __shared__ float sq2[256];           // this block's q^2
  __shared__ float sdist[8][32 * 16];  // per-wave 32x16 distance tile (transpose buf)

  const int tid  = threadIdx.x;
  const int b    = blockIdx.x >> 3;    // NN/256 == 8 blocks per batch
  const int qblk = blockIdx.x & 7;
  const int gq   = qblk * 256 + tid;   // this lane's query (global n index)

  const int tprev = (t == 0) ? 0 : (t - 1);
  const float* refp = pts + ((long)b * TT + tprev) * NN * 3;

  // 16-B vectorized ref staging: 6144 floats -> 1536 b128 transfers
  {
    const v4f* rp4 = (const v4f*)refp;
    v4f* sr4 = (v4f*)sref;
    #pragma unroll
    for (int i = 0; i < (NN * 3) / (4 * 256); ++i)
      sr4[i * 256 + tid] = rp4[i * 256 + tid];
  }

  const float qx = xcur[((long)b * NN + gq) * 3 + 0];
  const float qy = xcur[((long)b * NN + gq) * 3 + 1];
  const float qz = xcur[((long)b * NN + gq) * 3 + 2];
  sqc[tid * 3 + 0] = qx; sqc[tid * 3 + 1] = qy; sqc[tid * 3 + 2] = qz;
  sq2[tid] = qx * qx + qy * qy + qz * qz;
  __syncthreads();

  const int w  = tid >> 5;   // wave in block
  const int L  = tid & 31;   // lane
  const int m  = L & 15;
  const int hi = L >> 4;

  // A = -2 * Q for the wave's two 16-query M-tiles.
  // f32 16x4 A layout: VGPR0/1 = K0,K1 (lanes 0-15) / K2,K3 (lanes 16-31), M = lane%16.
  const int q0 = w * 32 + m;
  const int q1 = w * 32 + 16 + m;
  v2f a0, a1;
  if (hi == 0) {
    a0 = (v2f){-2.f * sqc[q0 * 3 + 0], -2.f * sqc[q0 * 3 + 1]};
    a1 = (v2f){-2.f * sqc[q1 * 3 + 0], -2.f * sqc[q1 * 3 + 1]};
  } else {
    a0 = (v2f){-2.f * sqc[q0 * 3 + 2], 0.f};   // K=3 pad
    a1 = (v2f){-2.f * sqc[q1 * 3 + 2], 0.f};
  }
  // C base = q2[row] (row layout: VGPR v -> M = (hi*8)+v, N = lane%16)
  v8f cb0, cb1;
  #pragma unroll
  for (int v = 0; v < 8; ++v) {
    cb0[v] = sq2[w * 32 + hi * 8 + v];
    cb1[v] = sq2[w * 32 + 16 + hi * 8 + v];
  }

  float bd[KNN]; int bix[KNN];
  #pragma unroll
  for (int i = 0; i < KNN; ++i) { bd[i] = 3.402823466e38f; bix[i] = 0; }

  float* wd = &sdist[w][0];
  for (int c = 0; c < NN / 16; ++c) {
    // B = ref coords, 4x16 (K x N): lane handles column n = lane%16.
    const int j = c * 16 + m;
    const float Rx = sref[j * 3 + 0], Ry = sref[j * 3 + 1], Rz = sref[j * 3 + 2];
    const float r2 = Rx * Rx + Ry * Ry + Rz * Rz;
    const v2f bv = (hi == 0) ? (v2f){Rx, Ry} : (v2f){Rz, 0.f};
    const v8f c0 = cb0 + r2;           // q2[m] + r2[n]
    const v8f c1 = cb1 + r2;
    // D = q2 + r2 - 2 q.r  (full squared-distance tile in one WMMA)
    v8f d0 = __builtin_amdgcn_wmma_f32_16x16x4_f32(false, a0, false, bv, (short)0, c0, false, false);
    v8f d1 = __builtin_amdgcn_wmma_f32_16x16x4_f32(false, a1, false, bv, (short)0, c1, false, false);

    // Transpose through per-wave LDS: column-per-lane -> row-per-lane.
    #pragma unroll
    for (int v = 0; v < 8; ++v) {
      wd[(hi * 8 + v) * 16 + m]      = d0[v];
      wd[(16 + hi * 8 + v) * 16 + m] = d1[v];
    }
    __builtin_amdgcn_wave_barrier();   // DS ops are in-order within the wave

    // Each lane owns one query row: pull it with 4 x ds_load_b128, then select
    // candidates entirely from registers.
    const v4f* row4 = (const v4f*)&wd[L * 16];   // 64-B aligned
    float cand[16];
    #pragma unroll
    for (int q = 0; q < 4; ++q) {
      const v4f rr = row4[q];
      cand[q * 4 + 0] = rr.x; cand[q * 4 + 1] = rr.y;
      cand[q * 4 + 2] = rr.z; cand[q * 4 + 3] = rr.w;
    }
    __builtin_amdgcn_wave_barrier();   // row consumed; next chunk may overwrite wd

    #pragma unroll
    for (int nn = 0; nn < 16; ++nn) {
      const float d = cand[nn];
      const int jj = c * 16 + nn;
      if (d < bd[KNN - 1]) {           // strict < keeps lax.top_k tie order
        bd[KNN - 1] = d; bix[KNN - 1] = jj;
        #pragma unroll
        for (int i = KNN - 1; i > 0; --i) {
          if (bd[i] < bd[i - 1]) {
            float td = bd[i]; bd[i] = bd[i - 1]; bd[i - 1] = td;
            int   ti = bix[i]; bix[i] = bix[i - 1]; bix[i - 1] = ti;
          }
        }
      }
    }
  }

  // ---- epilogue: gathers + all 7 outputs ----
  const float n0x = sref[bix[0] * 3 + 0];
  const float n0y = sref[bix[0] * 3 + 1];
  const float n0z = sref[bix[0] * 3 + 2];

  // anchor = patchlet_points[b, 0, gq, 0, :]
  float ax, ay, az;
  if (t == 0) { ax = n0x; ay = n0y; az = n0z; }
  else {
    const long ab = (((long)b * TT + 0) * NN + gq) * KNN * 3;
    ax = out[O_PTS + ab + 0]; ay = out[O_PTS + ab + 1]; az = out[O_PTS + ab + 2];
  }

  const long btn = ((long)b * TT + t) * NN + gq;

  // K-contiguous outputs as b128 stores (each lane owns 64 aligned bytes)
  {
    v4f* oidx = (v4f*)(out + O_IDX + btn * KNN);
    v4f* oplt = (v4f*)(out + O_PLT + btn * KNN);
    v4f* odst = (v4f*)(out + O_DST + btn * KNN);
    #pragma unroll
    for (int q = 0; q < 4; ++q) {
      const v4f vi = (v4f){(float)bix[q * 4 + 0], (float)bix[q * 4 + 1],
                           (float)bix[q * 4 + 2], (float)bix[q * 4 + 3]};
      const v4f vd = (v4f){bd[q * 4 + 0], bd[q * 4 + 1], bd[q * 4 + 2], bd[q * 4 + 3]};
      oidx[q] = vi;
      oplt[q] = vi;
      odst[q] = vd;
    }
  }

  #pragma unroll
  for (int kk = 0; kk < KNN; ++kk) {
    const float px = sref[bix[kk] * 3 + 0];
    const float py = sref[bix[kk] * 3 + 1];
    const float pz = sref[bix[kk] * 3 + 2];
    const long pb = (btn * KNN + kk) * 3;
    out[O_PTS + pb + 0] = px; out[O_PTS + pb + 1] = py; out[O_PTS + pb + 2] = pz;
    const float nx = px - ax, ny = py - ay, nz = pz - az;
    out[O_NRM + pb + 0] = nx; out[O_NRM + pb + 1] = ny; out[O_NRM + pb + 2] = nz;
    const long fb = (btn * KNN + kk) * 6;
    out[O_FTS + fb + 0] = px; out[O_FTS + fb + 1] = py; out[O_FTS + fb + 2] = pz;
    out[O_FTS + fb + 3] = nx; out[O_FTS + fb + 4] = ny; out[O_FTS + fb + 5] = nz;
  }
  const long ob = btn * 3;
  out[O_OUTX + ob + 0] = n0x; out[O_OUTX + ob + 1] = n0y; out[O_OUTX + ob + 2] = n0z;
  // x_cur for next step
  xcur[((long)b * NN + gq) * 3 + 0] = n0x;
  xcur[((long)b * NN + gq) * 3 + 1] = n0y;
  xcur[((long)b * NN + gq) * 3 + 2] = n0z;
}

extern "C" void kernel_launch(void* const* d_in, const int* in_sizes, int n_in,
                              void* d_out, int out_size, void* d_ws, size_t ws_size,
                              hipStream_t stream) {
  (void)in_sizes; (void)n_in; (void)out_size; (void)ws_size;
  const float* pts = (const float*)d_in[0];
  float* out  = (float*)d_out;
  float* xcur = (float*)d_ws;           // BB*NN*3 floats = 96 KB scratch

  patchlet_init<<<(BB * NN * 3 + 255) / 256, 256, 0, stream>>>(pts, xcur);
  for (int t = 0; t < TT; ++t) {
    patchlet_step<<<BB * (NN / 256), 256, 0, stream>>>(pts, xcur, out, t);
  }
}